// Feature_Distribution_Distance_func_49331994362322
// MI455X (gfx1250) — compile-verified
//
#include <hip/hip_runtime.h>
#include <hip/hip_bf16.h>
#include <stdint.h>

typedef float v2f __attribute__((ext_vector_type(2)));
typedef float v8f __attribute__((ext_vector_type(8)));

#define NB   32          // batch
#define NC   128         // channels
#define DPOOL 32768      // C*16*16 pooled feature length
#define NPIX 4096        // 64*64 per channel
#define POOL_WAVES 8

// ---------------------------------------------------------------------------
// Kernel 1: AvgPool2d(4,4) as Y = (P^T * X * P) / 16 using V_WMMA_F32_16X16X4_F32.
// One wave32 per (b,c) 64x64 image. X row-tiles staged into LDS with
// global_load_async_to_lds_b128 (contiguous 4KB per 16-row tile).
// ---------------------------------------------------------------------------
__global__ __launch_bounds__(256) void pool_wmma_kernel(
    const float* __restrict__ inA, const float* __restrict__ inB,
    float* __restrict__ fa, float* __restrict__ fb) {
  __shared__ float smem[POOL_WAVES * 2048];  // per wave: 1024 xtile + 1024 zbuf

  const int lane = threadIdx.x & 31;
  const int wave = threadIdx.x >> 5;
  const int ch   = blockIdx.x * POOL_WAVES + wave;   // 0..8191 (A then B)
  const bool isB = ch >= (NB * NC);
  const int  cc  = isB ? ch - NB * NC : ch;          // 0..4095
  const float* src = (isB ? inB : inA) + (size_t)cc * NPIX;
  const int b = cc >> 7;           // /128
  const int c = cc & 127;
  float* out = (isB ? fb : fa) + (size_t)b * DPOOL + (size_t)c * 256;

  float* xtile = smem + wave * 2048;
  float* zbuf  = xtile + 1024;     // Z = X*P, 64x16

  const int M  = lane & 15;        // A-matrix row / output row
  const int n  = lane & 15;        // B/C/D column
  const int kh = lane >> 4;        // lane half selects K offset +2 (ISA layout)

  // ---- Stage 1: Z(64x16) = X(64x64) * P(64x16), P[k][n] = (k>>2 == n) ----
  for (int t = 0; t < 4; ++t) {
    const float* gbase = src + t * 1024;             // 16 contiguous rows = 4KB
#pragma unroll
    for (int it = 0; it < 8; ++it) {
      const int idx16 = it * 32 + lane;              // 16-byte units
      unsigned ldsb = (unsigned)(uintptr_t)(xtile + idx16 * 4);
      const float* gp = gbase + idx16 * 4;
      asm volatile("global_load_async_to_lds_b128 %0, %1, off"
                   :: "v"(ldsb), "v"(gp) : "memory");
    }
    asm volatile("s_wait_asynccnt 0x0" ::: "memory");

    v8f acc = {};
#pragma unroll
    for (int s = 0; s < 16; ++s) {
      const int k0 = 4 * s + 2 * kh;                 // K index for vgpr0
      v2f a;                                         // A[M][k0], A[M][k0+1]
      a.x = xtile[M * 64 + k0];
      a.y = xtile[M * 64 + k0 + 1];
      v2f bm;                                        // P[k0][n], P[k0+1][n]
      bm.x = ((k0 >> 2) == n) ? 1.0f : 0.0f;
      bm.y = (((k0 + 1) >> 2) == n) ? 1.0f : 0.0f;
      acc = __builtin_amdgcn_wmma_f32_16x16x4_f32(false, a, false, bm,
                                                  (short)0, acc, false, false);
    }
#pragma unroll
    for (int r = 0; r < 8; ++r) {                    // C/D: vgpr r -> row r (+8 hi half)
      const int row = t * 16 + r + kh * 8;
      zbuf[row * 16 + n] = acc[r];
    }
  }

  // ---- Stage 2: Y(16x16) = P^T(16x64) * Z(64x16) ----
  v8f acc2 = {};
#pragma unroll
  for (int s = 0; s < 16; ++s) {
    const int k0 = 4 * s + 2 * kh;
    v2f a;                                           // P^T[M][k] = (k>>2 == M)
    a.x = ((k0 >> 2) == M) ? 1.0f : 0.0f;
    a.y = (((k0 + 1) >> 2) == M) ? 1.0f : 0.0f;
    v2f bm;                                          // Z[k][n]
    bm.x = zbuf[k0 * 16 + n];
    bm.y = zbuf[(k0 + 1) * 16 + n];
    acc2 = __builtin_amdgcn_wmma_f32_16x16x4_f32(false, a, false, bm,
                                                 (short)0, acc2, false, false);
  }
#pragma unroll
  for (int r = 0; r < 8; ++r) {
    const int m = r + kh * 8;
    out[m * 16 + n] = acc2[r] * 0.0625f;             // /16 average
  }
}

// ---------------------------------------------------------------------------
// Kernel 2: per-pair partial sums of sin(T*d)/d (d==0 -> T).
// blockIdx.y: 0=xx, 1=yy, 2=xy.  blockIdx.x: pair (i*32+j). 256 threads/pair.
// ---------------------------------------------------------------------------
__global__ __launch_bounds__(256) void pair_kernel(
    const float* __restrict__ fa, const float* __restrict__ fb,
    float* __restrict__ partials) {
  const int z    = blockIdx.y;
  const int pair = blockIdx.x;
  const int i = pair >> 5;
  const int j = pair & 31;
  const float* ar = ((z == 1) ? fb : fa) + (size_t)i * DPOOL;
  const float* br = ((z == 0) ? fa : fb) + (size_t)j * DPOOL;

  float s = 0.0f;
  for (int d0 = threadIdx.x * 4; d0 < DPOOL; d0 += 256 * 4) {
    const float4 av = *(const float4*)(ar + d0);
    const float4 bv = *(const float4*)(br + d0);
    const float dd[4] = {av.x - bv.x, av.y - bv.y, av.z - bv.z, av.w - bv.w};
#pragma unroll
    for (int q = 0; q < 4; ++q) {
      const float d  = dd[q];
      const bool  zr = (d == 0.0f);
      const float num = zr ? 100.0f : __sinf(d * 100.0f);
      const float den = zr ? 1.0f : d;
      s += __fdividef(num, den);
    }
  }

  __shared__ float red[256];
  red[threadIdx.x] = s;
  __syncthreads();
  for (int st = 128; st > 0; st >>= 1) {
    if (threadIdx.x < (unsigned)st) red[threadIdx.x] += red[threadIdx.x + st];
    __syncthreads();
  }
  if (threadIdx.x == 0) partials[z * 1024 + pair] = red[0];
}

// ---------------------------------------------------------------------------
// Kernel 3: reduce 3x1024 partials -> (Sxx + Syy - 2*Sxy) / (32*32*32768)
// ---------------------------------------------------------------------------
__global__ __launch_bounds__(256) void finalize_kernel(
    const float* __restrict__ partials, float* __restrict__ out) {
  __shared__ float red[3][256];
  float s0 = 0.f, s1 = 0.f, s2 = 0.f;
  for (int t = threadIdx.x; t < 1024; t += 256) {
    s0 += partials[t];
    s1 += partials[1024 + t];
    s2 += partials[2048 + t];
  }
  red[0][threadIdx.x] = s0;
  red[1][threadIdx.x] = s1;
  red[2][threadIdx.x] = s2;
  __syncthreads();
  for (int st = 128; st > 0; st >>= 1) {
    if (threadIdx.x < (unsigned)st) {
      red[0][threadIdx.x] += red[0][threadIdx.x + st];
      red[1][threadIdx.x] += red[1][threadIdx.x + st];
      red[2][threadIdx.x] += red[2][threadIdx.x + st];
    }
    __syncthreads();
  }
  if (threadIdx.x == 0)
    out[0] = (red[0][0] + red[1][0] - 2.0f * red[2][0]) * (1.0f / 33554432.0f);
}

extern "C" void kernel_launch(void* const* d_in, const int* in_sizes, int n_in,
                              void* d_out, int out_size, void* d_ws, size_t ws_size,
                              hipStream_t stream) {
  const float* inA = (const float*)d_in[0];
  const float* inB = (const float*)d_in[1];
  float* fa       = (float*)d_ws;                  // 32*32768 floats (4 MB)
  float* fb       = fa + (size_t)NB * DPOOL;       // 4 MB
  float* partials = fb + (size_t)NB * DPOOL;       // 3*1024 floats
  float* out      = (float*)d_out;

  // 8192 channel-images, 8 waves (one image each) per block
  pool_wmma_kernel<<<dim3(8192 / POOL_WAVES), dim3(256), 0, stream>>>(inA, inB, fa, fb);
  pair_kernel<<<dim3(1024, 3, 1), dim3(256), 0, stream>>>(fa, fb, partials);
  finalize_kernel<<<dim3(1), dim3(256), 0, stream>>>(partials, out);
}